// SignedGIN_9852654977716
// MI455X (gfx1250) — compile-verified
//
#include <hip/hip_runtime.h>

typedef float v2f __attribute__((ext_vector_type(2)));
typedef float v8f __attribute__((ext_vector_type(8)));

#define NN 50000
#define EE 800000
#define DD 256     // hidden dim
#define DN 512     // raw node dim
#define DE 64      // raw edge dim

// D = A(16x4 f32) * B(4x16 f32) + C(16x16 f32)
__device__ __forceinline__ v8f wmma4(v2f a, v2f b, v8f c) {
    return __builtin_amdgcn_wmma_f32_16x16x4_f32(
        /*neg_a=*/false, a, /*neg_b=*/false, b,
        /*c_mod=*/(short)0, c, /*reuse_a=*/false, /*reuse_b=*/false);
}

// ---------------------------------------------------------------------------
// Generic node GEMM: out[M, 256] = act(X[M,K] @ W[256,K]^T + bias)
// One wave computes a 16-row x 256-col stripe (16 WMMA accumulators).
// ---------------------------------------------------------------------------
__global__ __launch_bounds__(256)
void k_gemm(const float* __restrict__ X, const float* __restrict__ W,
            const float* __restrict__ bias, float* __restrict__ out,
            int M, int K, int do_relu)
{
    const int wave   = threadIdx.x >> 5;
    const int tile   = blockIdx.x * (blockDim.x >> 5) + wave;
    const int ntiles = M >> 4;
    if (tile >= ntiles) return;

    const int lane = threadIdx.x & 31;
    const int half = lane >> 4;     // 0: K 0..1 / M j ; 1: K 2..3 / M j+8
    const int lm   = lane & 15;
    const int kb   = half << 1;
    const int m0   = tile << 4;

    v8f acc[16];
#pragma unroll
    for (int nt = 0; nt < 16; ++nt) {
        const float bv = bias[nt * 16 + lm];
        acc[nt] = (v8f){bv, bv, bv, bv, bv, bv, bv, bv};
    }

    const float* xrow = X + (size_t)(m0 + lm) * K;   // A row M = lm (both halves)
    for (int k = 0; k < K; k += 4) {
        const v2f a = *(const v2f*)(xrow + k + kb);  // K = k+kb, k+kb+1
#pragma unroll
        for (int nt = 0; nt < 16; ++nt) {
            const v2f b = *(const v2f*)(W + (size_t)(nt * 16 + lm) * K + k + kb);
            acc[nt] = wmma4(a, b, acc[nt]);
        }
    }

#pragma unroll
    for (int nt = 0; nt < 16; ++nt) {
#pragma unroll
        for (int j = 0; j < 8; ++j) {
            const int m = m0 + j + half * 8;         // C layout: VGPR j -> M=j(+8)
            float v = acc[nt][j];
            if (do_relu) v = fmaxf(v, 0.0f);
            out[(size_t)m * DD + nt * 16 + lm] = v;
        }
    }
}

// ---------------------------------------------------------------------------
// Fused edge pass: for a 16-edge tile
//   e = ef @ We^T + be               (WMMA, in registers)
//   msg = (h[src] + e) * w/wsum[dst] (fused gather)
//   acc[dst] += msg                  (global fp32 atomics)
// ---------------------------------------------------------------------------
__global__ __launch_bounds__(256)
void k_edge(const float* __restrict__ ef, const float* __restrict__ We,
            const float* __restrict__ be, const float* __restrict__ w,
            const float* __restrict__ wsum, const int* __restrict__ src,
            const int* __restrict__ dst, const float* __restrict__ h,
            float* __restrict__ accbuf)
{
    const int wave = threadIdx.x >> 5;
    const int tile = blockIdx.x * (blockDim.x >> 5) + wave;
    if (tile >= (EE >> 4)) return;

    const int lane = threadIdx.x & 31;
    const int half = lane >> 4;
    const int lm   = lane & 15;
    const int kb   = half << 1;
    const int e0   = tile << 4;

    v8f acc[16];
#pragma unroll
    for (int nt = 0; nt < 16; ++nt) {
        const float bv = be[nt * 16 + lm];
        acc[nt] = (v8f){bv, bv, bv, bv, bv, bv, bv, bv};
    }

    const float* arow = ef + (size_t)(e0 + lm) * DE;
#pragma unroll
    for (int k = 0; k < DE; k += 4) {
        const v2f a = *(const v2f*)(arow + k + kb);
#pragma unroll
        for (int nt = 0; nt < 16; ++nt) {
            const v2f b = *(const v2f*)(We + (size_t)(nt * 16 + lm) * DE + k + kb);
            acc[nt] = wmma4(a, b, acc[nt]);
        }
    }

    // Epilogue: each lane owns (M = j + 8*half, N = nt*16 + lm)
#pragma unroll
    for (int j = 0; j < 8; ++j) {
        const int   eidx = e0 + j + half * 8;
        const int   s    = src[eidx];
        const int   d    = dst[eidx];
        const float wn   = w[eidx] / wsum[d];        // EdgeWeightNorm 'right'
        const float* hrow = h + (size_t)s * DD;
        float*       orow = accbuf + (size_t)d * DD;
#pragma unroll
        for (int nt = 0; nt < 16; ++nt) {
            const int   c  = nt * 16 + lm;
            const float mv = (hrow[c] + acc[nt][j]) * wn;
            atomicAdd(orow + c, mv);
        }
    }
}

// ---------------------------------------------------------------------------
__global__ void k_wsum(const float* __restrict__ w, const int* __restrict__ dst,
                       float* __restrict__ wsum)
{
    const int i = blockIdx.x * blockDim.x + threadIdx.x;
    if (i < EE) atomicAdd(&wsum[dst[i]], w[i]);
}

__global__ void k_deg(const int* __restrict__ dst, float* __restrict__ deg)
{
    const int i = blockIdx.x * blockDim.x + threadIdx.x;
    if (i < EE) atomicAdd(&deg[dst[i]], 1.0f);
}

// rst = h + acc / max(deg, 1)
__global__ void k_finalize(const float* __restrict__ h, const float* __restrict__ acc,
                           const float* __restrict__ deg, float* __restrict__ rst)
{
    const int i = blockIdx.x * blockDim.x + threadIdx.x;
    if (i < NN * DD) {
        const int node = i >> 8;                      // DD == 256
        const float dg = fmaxf(deg[node], 1.0f);
        rst[i] = h[i] + acc[i] / dg;
    }
}

// ---------------------------------------------------------------------------
extern "C" void kernel_launch(void* const* d_in, const int* /*in_sizes*/, int /*n_in*/,
                              void* d_out, int /*out_size*/, void* d_ws, size_t /*ws_size*/,
                              hipStream_t stream)
{
    const float* node_feats = (const float*)d_in[0];
    const float* edge_feats = (const float*)d_in[1];  // [3, E, 64]
    const float* edge_w     = (const float*)d_in[2];  // [3, E]
    const int*   src        = (const int*)d_in[3];
    const int*   dst        = (const int*)d_in[4];
    const float* Wn = (const float*)d_in[5];
    const float* bn = (const float*)d_in[6];
    const float* We = (const float*)d_in[7];
    const float* be = (const float*)d_in[8];
    const float* W0 = (const float*)d_in[9];
    const float* b0 = (const float*)d_in[10];
    const float* W2 = (const float*)d_in[11];
    const float* b2 = (const float*)d_in[12];
    float* out = (float*)d_out;

    // Workspace layout (floats)
    float* h    = (float*)d_ws;                       // [N, 256]
    float* rst  = h    + (size_t)NN * DD;             // [N, 256]
    float* accb = rst  + (size_t)NN * DD;             // [N, 256]
    float* wsum = accb + (size_t)NN * DD;             // [N]
    float* deg  = wsum + NN;                          // [N]

    const int gemm_blocks = ((NN / 16) + 7) / 8;      // 8 waves/block
    const int edge_blocks = ((EE / 16) + 7) / 8;
    const int ew_blocks   = (EE + 255) / 256;
    const int fin_blocks  = (NN * DD + 255) / 256;

    // deg (shared by all layers)
    hipMemsetAsync(deg, 0, NN * sizeof(float), stream);
    k_deg<<<ew_blocks, 256, 0, stream>>>(dst, deg);

    // h = node_feats @ Wn^T + bn
    k_gemm<<<gemm_blocks, 256, 0, stream>>>(node_feats, Wn, bn, h, NN, DN, 0);

    for (int l = 0; l < 3; ++l) {
        const float* ef = edge_feats + (size_t)l * EE * DE;
        const float* w  = edge_w + (size_t)l * EE;
        const float* Wl = (l == 2) ? W2 : W0;
        const float* bl = (l == 2) ? b2 : b0;
        float* hout     = (l == 2) ? out : h;

        hipMemsetAsync(wsum, 0, NN * sizeof(float), stream);
        hipMemsetAsync(accb, 0, (size_t)NN * DD * sizeof(float), stream);
        k_wsum<<<ew_blocks, 256, 0, stream>>>(w, dst, wsum);
        k_edge<<<edge_blocks, 256, 0, stream>>>(ef, We, be, w, wsum, src, dst, h, accb);
        k_finalize<<<fin_blocks, 256, 0, stream>>>(h, accb, deg, rst);
        k_gemm<<<gemm_blocks, 256, 0, stream>>>(rst, Wl, bl, hout, NN, DD, 1);
    }
}